// RowMaskHandler_50921132261430
// MI455X (gfx1250) — compile-verified
//
#include <hip/hip_runtime.h>
#include <hip/hip_bf16.h>
#include <math.h>

// Problem shape (fixed by the reference's setup_inputs)
#define B_DIM 8
#define R_DIM 4096
#define C_DIM 1024
#define NROWS (B_DIM * R_DIM)   // 32768

typedef __attribute__((ext_vector_type(2))) float v2f;
typedef __attribute__((ext_vector_type(4))) float v4f;
typedef __attribute__((ext_vector_type(8))) float v8f;

// ---------------------------------------------------------------------------
// Kernel 1: per-row L2 norm. One 256-thread block per row; each thread loads
// one v4f (global_load_b128), squares, LDS tree-reduce, sqrt.
// These loads are the first touch of W -> default RT hint populates L2.
// ---------------------------------------------------------------------------
__global__ __launch_bounds__(256) void rowmag_kernel(const v4f* __restrict__ w,
                                                     float* __restrict__ mags) {
    __shared__ float red[256];
    const int row = blockIdx.x;                        // 0..32767
    const v4f* p = w + (size_t)row * (C_DIM / 4);
    v4f v = p[threadIdx.x];
    float s = v.x * v.x + v.y * v.y + v.z * v.z + v.w * v.w;
    red[threadIdx.x] = s;
    __syncthreads();
    for (int off = 128; off > 0; off >>= 1) {
        if (threadIdx.x < off) red[threadIdx.x] += red[threadIdx.x + off];
        __syncthreads();
    }
    if (threadIdx.x == 0) mags[row] = sqrtf(red[0]);
}

// ---------------------------------------------------------------------------
// Kernel 2: score = logits[8,1024] @ score_w[1024,1] via V_WMMA_F32_16X16X4_F32.
// Single wave32, fully branchless: every lane loads a VALID (clamped) address
// and zeroes its contribution by multiplying with a 0/1 lane mask, so no
// exec-mask branches are emitted and EXEC is all-ones at every WMMA.
//
// CDNA5 16x4 f32 A-layout: lanes 0-15 hold A[m][K=0],A[m][K=1];
// lanes 16-31 hold A[m][K=2],A[m][K=3]. B (4x16): lane n = column n, only
// column 0 carries score_w. D: column 0 = lane 0, VGPRs 0..7 = scores b=0..7.
// ---------------------------------------------------------------------------
__global__ __launch_bounds__(32) void score_kernel(const float* __restrict__ logits,
                                                   const float* __restrict__ score_w,
                                                   const float* __restrict__ score_b,
                                                   int* __restrict__ kcount) {
    const int lane = threadIdx.x;        // 0..31
    const int half = lane >> 4;          // 0 => K 0,1 ; 1 => K 2,3
    const int m = lane & 15;             // matrix row (batch index when < 8)
    const float mvF = (m < B_DIM) ? 1.0f : 0.0f;   // valid A rows
    const float bvF = (m == 0) ? 1.0f : 0.0f;      // only N=0 column of B used
    const int mc = (m < B_DIM) ? m : 0;            // clamped (always-valid) row

    const float* __restrict__ lp = logits + mc * C_DIM;

    v8f acc = {};
    for (int kc = 0; kc < C_DIM; kc += 4) {
        const int k0 = kc + half * 2;
        const int k1 = k0 + 1;
        float a0 = lp[k0] * mvF;
        float a1 = lp[k1] * mvF;
        float b0 = score_w[k0] * bvF;
        float b1 = score_w[k1] * bvF;
        v2f a = {a0, a1};
        v2f b = {b0, b1};
        acc = __builtin_amdgcn_wmma_f32_16x16x4_f32(
            /*neg_a=*/false, a, /*neg_b=*/false, b,
            /*c_mod=*/(short)0, acc, /*reuse_a=*/false, /*reuse_b=*/false);
    }

    if (lane == 0) {
        const float bias = score_b[0];
        for (int bb = 0; bb < B_DIM; ++bb) {
            float s = acc[bb] + bias;
            float kf = 1.0f / (1.0f + expf(-s));        // sigmoid
            int k = (int)(kf * (float)R_DIM);           // truncate like astype(int32)
            if (k < 1) k = 1;
            if (k > R_DIM) k = R_DIM;
            kcount[bb] = k;
        }
    }
}

// ---------------------------------------------------------------------------
// Kernel 3: per-batch bitonic sort (ascending) of 4096 row norms in LDS;
// threshold = k-th largest = sorted_asc[4096 - k]. 8 blocks x 1024 threads.
// ---------------------------------------------------------------------------
__global__ __launch_bounds__(1024) void thresh_kernel(const float* __restrict__ mags,
                                                      const int* __restrict__ kcount,
                                                      float* __restrict__ thresh) {
    __shared__ float s[R_DIM];
    const int b = blockIdx.x;
    const int tid = threadIdx.x;
    for (int i = tid; i < R_DIM; i += 1024) s[i] = mags[b * R_DIM + i];

    for (int k = 2; k <= R_DIM; k <<= 1) {
        for (int j = k >> 1; j > 0; j >>= 1) {
            __syncthreads();
            for (int i = tid; i < R_DIM; i += 1024) {
                const int ixj = i ^ j;
                if (ixj > i) {
                    float a = s[i], c = s[ixj];
                    const bool up = ((i & k) == 0);
                    if ((a > c) == up) { s[i] = c; s[ixj] = a; }
                }
            }
        }
    }
    __syncthreads();
    if (tid == 0) thresh[b] = s[R_DIM - kcount[b]];
}

// ---------------------------------------------------------------------------
// Kernel 4: apply mask. One block per row; keep/drop is uniform across the
// block (no divergence). Dropped rows write zeros WITHOUT reading the source.
// Output stores are NON-TEMPORAL (TH=NT on global_store_b128) so the 134MB of
// writes do not evict W from the 192MB L2 -> kept-row reads mostly hit L2
// (populated by kernel 1).
// ---------------------------------------------------------------------------
__global__ __launch_bounds__(256) void mask_kernel(const v4f* __restrict__ w,
                                                   const float* __restrict__ mags,
                                                   const float* __restrict__ thresh,
                                                   v4f* __restrict__ out) {
    const int row = blockIdx.x;
    const int b = row >> 12;                     // row / 4096
    const bool keep = mags[row] >= thresh[b];
    const size_t idx = (size_t)row * (C_DIM / 4) + threadIdx.x;
    v4f val;
    if (keep) {
        val = w[idx];
    } else {
        val = (v4f){0.0f, 0.0f, 0.0f, 0.0f};
    }
    __builtin_nontemporal_store(val, &out[idx]);
}

// ---------------------------------------------------------------------------
extern "C" void kernel_launch(void* const* d_in, const int* in_sizes, int n_in,
                              void* d_out, int out_size, void* d_ws, size_t ws_size,
                              hipStream_t stream) {
    const float* w       = (const float*)d_in[0];   // [8,4096,1024]
    const float* logits  = (const float*)d_in[1];   // [8,1024]
    const float* score_w = (const float*)d_in[2];   // [1024,1]
    const float* score_b = (const float*)d_in[3];   // [1]
    v4f* out = (v4f*)d_out;

    // Workspace layout: 32768 f32 row norms | 8 int32 k | 8 f32 thresholds
    float* mags   = (float*)d_ws;
    int*   kcount = (int*)((char*)d_ws + (size_t)NROWS * sizeof(float));
    float* thresh = (float*)((char*)d_ws + (size_t)NROWS * sizeof(float)
                                         + B_DIM * sizeof(int));

    rowmag_kernel<<<NROWS, 256, 0, stream>>>((const v4f*)w, mags);
    score_kernel<<<1, 32, 0, stream>>>(logits, score_w, score_b, kcount);
    thresh_kernel<<<B_DIM, 1024, 0, stream>>>(mags, kcount, thresh);
    mask_kernel<<<NROWS, 256, 0, stream>>>((const v4f*)w, mags, thresh, out);
}